// OSG_C_25202868092938
// MI455X (gfx1250) — compile-verified
//
#include <hip/hip_runtime.h>
#include <hip/hip_bf16.h>

// ---------------------------------------------------------------------------
// Sizes fixed by the reference: b=8, N=256, f0=512, f1=256, f2=128, K_MAX=30
// ---------------------------------------------------------------------------
#define NB    8
#define NSEQ  256
#define F0    512
#define F1    256
#define F2    128
#define MTOT  (NB * NSEQ)   // 2048
#define KMAX  30

typedef __attribute__((ext_vector_type(2))) float v2f;
typedef __attribute__((ext_vector_type(8))) float v8f;

// ---------------------------------------------------------------------------
// Wave-level fp32 WMMA 16x16 tile:  C(16x16) += A(16xK) * B(Kx16)
// A row-major (lda), B row-major (ldb).  Per ISA layout for
// V_WMMA_F32_16X16X4_F32:
//   A frag: lane l<16 holds A[l][k], A[l][k+1]; lane l>=16 holds A[l-16][k+2..3]
//   B frag: lane l<16 holds B[k][l], B[k+1][l]; lane l>=16 holds B[k+2..3][l-16]
//   C/D   : vgpr v, lane l<16 -> (M=v,   N=l); lane l>=16 -> (M=v+8, N=l-16)
// ---------------------------------------------------------------------------
template <int K, int LDA, int LDB>
__device__ __forceinline__ v8f wmma_tile(const float* __restrict__ A,
                                         const float* __restrict__ B,
                                         int lane) {
  const int half = lane >> 4;   // 0 or 1
  const int l15  = lane & 15;
  v8f c = {};
  const float* ap = A + (size_t)l15 * LDA + 2 * half;   // contiguous float2
  const float* bp = B + (size_t)(2 * half) * LDB + l15; // strided pair
#pragma unroll 8
  for (int k = 0; k < K; k += 4) {
    v2f a = *(const v2f*)(ap + k);
    v2f b;
    b.x = bp[(size_t)k * LDB];
    b.y = bp[(size_t)k * LDB + LDB];
    c = __builtin_amdgcn_wmma_f32_16x16x4_f32(false, a, false, b,
                                              (short)0, c, false, false);
  }
  return c;
}

// Same but B given as Bt (N x K row-major), i.e. computes A * Bt^T.
// Fragment loads for both operands become contiguous float2.
template <int K, int LDA>
__device__ __forceinline__ v8f wmma_tile_bt(const float* __restrict__ A,
                                            const float* __restrict__ Bt,
                                            int lane) {
  const int half = lane >> 4;
  const int l15  = lane & 15;
  v8f c = {};
  const float* ap = A  + (size_t)l15 * LDA + 2 * half;
  const float* bp = Bt + (size_t)l15 * K   + 2 * half;
#pragma unroll 8
  for (int k = 0; k < K; k += 4) {
    v2f a = *(const v2f*)(ap + k);
    v2f b = *(const v2f*)(bp + k);
    c = __builtin_amdgcn_wmma_f32_16x16x4_f32(false, a, false, b,
                                              (short)0, c, false, false);
  }
  return c;
}

// ---------------------------------------------------------------------------
// Kernel 1: h = relu(x @ W0 + b0)    (2048x512)@(512x256)
// 4 waves per block, one 16x16 output tile per wave.
// ---------------------------------------------------------------------------
__global__ void k_mlp1(const float* __restrict__ x, const float* __restrict__ W0,
                       const float* __restrict__ b0, float* __restrict__ h) {
  const int lane = threadIdx.x & 31;
  const int wave = threadIdx.x >> 5;
  const int tile = blockIdx.x * 4 + wave;     // 2048 tiles
  const int mt = tile >> 4;                   // 128 M-tiles
  const int nt = tile & 15;                   // 16  N-tiles
  v8f c = wmma_tile<F0, F0, F1>(x + (size_t)(mt * 16) * F0, W0 + nt * 16, lane);
  const int half = lane >> 4, l15 = lane & 15;
  const int col = nt * 16 + l15;
  const float bias = b0[col];
#pragma unroll
  for (int v = 0; v < 8; ++v) {
    const int row = mt * 16 + half * 8 + v;
    float val = c[v] + bias;
    h[(size_t)row * F1 + col] = val > 0.f ? val : 0.f;
  }
}

// ---------------------------------------------------------------------------
// Kernel 2: z = h @ W1 + b1          (2048x256)@(256x128)
// ---------------------------------------------------------------------------
__global__ void k_mlp2(const float* __restrict__ h, const float* __restrict__ W1,
                       const float* __restrict__ b1, float* __restrict__ z) {
  const int lane = threadIdx.x & 31;
  const int wave = threadIdx.x >> 5;
  const int tile = blockIdx.x * 4 + wave;     // 1024 tiles
  const int mt = tile >> 3;                   // 128 M-tiles
  const int nt = tile & 7;                    // 8   N-tiles
  v8f c = wmma_tile<F1, F1, F2>(h + (size_t)(mt * 16) * F1, W1 + nt * 16, lane);
  const int half = lane >> 4, l15 = lane & 15;
  const int col = nt * 16 + l15;
  const float bias = b1[col];
#pragma unroll
  for (int v = 0; v < 8; ++v) {
    const int row = mt * 16 + half * 8 + v;
    z[(size_t)row * F2 + col] = c[v] + bias;
  }
}

// ---------------------------------------------------------------------------
// Kernel 3: corr[b] = z_b @ z_b^T    per batch (256x128)@(128x256)
// B operand is z^T, so fragments are contiguous in k -> wmma_tile_bt.
// ---------------------------------------------------------------------------
__global__ void k_corr(const float* __restrict__ z, float* __restrict__ corr) {
  const int lane = threadIdx.x & 31;
  const int wave = threadIdx.x >> 5;
  const int tile = blockIdx.x * 4 + wave;     // 2048 tiles
  const int b  = tile >> 8;
  const int mt = (tile >> 4) & 15;
  const int nt = tile & 15;
  const float* zb = z + (size_t)b * NSEQ * F2;
  v8f c = wmma_tile_bt<F2, F2>(zb + (size_t)(mt * 16) * F2,
                               zb + (size_t)(nt * 16) * F2, lane);
  const int half = lane >> 4, l15 = lane & 15;
  float* cb = corr + (size_t)b * NSEQ * NSEQ;
#pragma unroll
  for (int v = 0; v < 8; ++v) {
    const int row = mt * 16 + half * 8 + v;
    cb[(size_t)row * NSEQ + nt * 16 + l15] = c[v];
  }
}

// ---------------------------------------------------------------------------
// Kernel 4: D = (1 - corr / sqrt(max(d_r*d_c, 1e-8))) / 2
// ---------------------------------------------------------------------------
__global__ void k_dist(const float* __restrict__ corr, float* __restrict__ D) {
  const int idx = blockIdx.x * 256 + threadIdx.x;   // 524288 total
  const int b = idx >> 16, r = (idx >> 8) & 255, c = idx & 255;
  const float* cb = corr + (size_t)b * NSEQ * NSEQ;
  const float dr = cb[r * NSEQ + r];
  const float dc = cb[c * NSEQ + c];
  const float den = sqrtf(fmaxf(dr * dc, 1e-8f));
  D[idx] = 0.5f * (1.0f - cb[r * NSEQ + c] / den);
}

// ---------------------------------------------------------------------------
// Kernels 5/6: 2-D inclusive prefix sum P of D (per batch), in place.
// ---------------------------------------------------------------------------
__global__ void k_colsum(const float* __restrict__ D, float* __restrict__ P) {
  const int b = blockIdx.x, c = threadIdx.x;
  const float* Db = D + (size_t)b * NSEQ * NSEQ;
  float* Pb = P + (size_t)b * NSEQ * NSEQ;
  float acc = 0.f;
  for (int r = 0; r < NSEQ; ++r) {          // coalesced across threads
    acc += Db[r * NSEQ + c];
    Pb[r * NSEQ + c] = acc;
  }
}

__global__ void k_rowsum(float* __restrict__ P) {
  const int b = blockIdx.x, r = threadIdx.x;
  float* row = P + (size_t)b * NSEQ * NSEQ + (size_t)r * NSEQ;
  float acc = 0.f;
  for (int c = 0; c < NSEQ; ++c) { acc += row[c]; row[c] = acc; }
}

// ---------------------------------------------------------------------------
// Kernel 7: Ds[r][c] = D[r][c] if r-c==1, else block sum over [lo..hi]^2
// ---------------------------------------------------------------------------
__global__ void k_dsum(const float* __restrict__ D, const float* __restrict__ P,
                       float* __restrict__ Ds) {
  const int idx = blockIdx.x * 256 + threadIdx.x;
  const int b = idx >> 16, r = (idx >> 8) & 255, c = idx & 255;
  const float* Pb = P + (size_t)b * NSEQ * NSEQ;
  const int lo = r < c ? r : c;
  const int hi = r < c ? c : r;
  auto pp = [&](int i, int j) -> float {
    return (i > 0 && j > 0) ? Pb[(i - 1) * NSEQ + (j - 1)] : 0.f;
  };
  const float blk = pp(hi + 1, hi + 1) - pp(lo, hi + 1)
                  - pp(hi + 1, lo)     + pp(lo, lo);
  Ds[idx] = (r - c == 1) ? D[idx] : blk;
}

// ---------------------------------------------------------------------------
// Kernel 8: sequential DP over kk = 1..29 + fused output reduction.
// One block per batch, 8 waves; each wave owns 32 rows per step.
// softmax(-T) max-pass == -min(T) == -Cmin, so one reduction serves both.
// ---------------------------------------------------------------------------
__global__ void k_dp(const float* __restrict__ DsAll, float* __restrict__ out) {
  __shared__ float s_cp[2][NSEQ + 1];     // double-buffered C_prev (+pad slot)
  __shared__ float s_acc[8][NSEQ];        // per-wave softmax accumulators
  const int b    = blockIdx.x;
  const int t    = threadIdx.x;
  const int wave = t >> 5;
  const int lane = t & 31;
  const float* Ds = DsAll + (size_t)b * NSEQ * NSEQ;

  s_cp[0][t] = Ds[t * NSEQ + (NSEQ - 1)];           // C_prev[n] = Ds[n][N-1]
  if (t == 0) { s_cp[0][NSEQ] = 0.f; s_cp[1][NSEQ] = 0.f; }
#pragma unroll
  for (int w = 0; w < 8; ++w) s_acc[w][t] = 0.f;
  __syncthreads();

  int rd = 0;
  for (int kk = 1; kk < KMAX; ++kk) {
    const int wr  = rd ^ 1;
    const int lim = NSEQ - kk;                       // valid: n<=i<lim
    for (int j = 0; j < 32; ++j) {
      const int n = wave + 8 * j;                    // rows 0..255 covered
      float tv[8];
      float tmin = 1e30f;
#pragma unroll
      for (int u = 0; u < 8; ++u) {
        const int i = lane + 32 * u;
        const float T = Ds[n * NSEQ + i] + s_cp[rd][i + 1];   // Cshift
        const bool valid = (i >= n) && (i < lim);
        tv[u] = valid ? T : 1e30f;
        tmin  = fminf(tmin, tv[u]);
      }
#pragma unroll
      for (int off = 16; off > 0; off >>= 1)
        tmin = fminf(tmin, __shfl_xor(tmin, off, 32));
      if (n < lim) {                                 // row has valid entries
        float ev[8], se = 0.f;
#pragma unroll
        for (int u = 0; u < 8; ++u) {
          const float e = (tv[u] < 1e29f) ? __expf(tmin - tv[u]) : 0.f;
          ev[u] = e; se += e;
        }
#pragma unroll
        for (int off = 16; off > 0; off >>= 1)
          se += __shfl_xor(se, off, 32);
        const float inv = 1.f / se;
#pragma unroll
        for (int u = 0; u < 8; ++u)
          if (ev[u] > 0.f) s_acc[wave][lane + 32 * u] += ev[u] * inv;
        if (lane == 0) s_cp[wr][n] = tmin;           // Cmin
      } else {
        if (lane == 0) s_cp[wr][n] = 0.f;            // n >= N-kk -> 0
      }
    }
    __syncthreads();
    rd ^= 1;
  }

  // Output:  out[b][i] = (sum sm + CA0 term) / mask count
  float num = 0.f;
#pragma unroll
  for (int w = 0; w < 8; ++w) num += s_acc[w][t];
  if (t == NSEQ - 1) num += (float)NSEQ;             // kk=0 plane: col N-1 = 1
  int cnt = (NSEQ - 1) - t;                          // #valid kk in 1..29
  cnt = cnt > 29 ? 29 : (cnt < 0 ? 0 : cnt);
  const float den = (t == NSEQ - 1 ? (float)NSEQ : 0.f) + (float)(t + 1) * (float)cnt;
  out[b * NSEQ + t] = num / den;
}

// ---------------------------------------------------------------------------
// Host launcher.  Workspace layout (floats), ~7 MB total:
//   [0            , 524288): h        -> reused as Ds
//   [524288       , 786432): z
//   [786432       ,1310720): corr     -> reused as P (prefix sums)
//   [1310720      ,1835008): D
// ---------------------------------------------------------------------------
extern "C" void kernel_launch(void* const* d_in, const int* in_sizes, int n_in,
                              void* d_out, int out_size, void* d_ws, size_t ws_size,
                              hipStream_t stream) {
  const float* x  = (const float*)d_in[0];
  const float* W0 = (const float*)d_in[1];
  const float* b0 = (const float*)d_in[2];
  const float* W1 = (const float*)d_in[3];
  const float* b1 = (const float*)d_in[4];
  float* ws   = (float*)d_ws;
  float* h    = ws;                       // 2048*256
  float* z    = ws + 524288;              // 2048*128
  float* corr = ws + 786432;              // 8*256*256
  float* D    = ws + 1310720;             // 8*256*256
  float* Ds   = h;                        // alias: h is dead after k_mlp2
  float* P    = corr;                     // alias: corr dead after k_dist

  k_mlp1 <<<512,  128, 0, stream>>>(x, W0, b0, h);
  k_mlp2 <<<256,  128, 0, stream>>>(h, W1, b1, z);
  k_corr <<<512,  128, 0, stream>>>(z, corr);
  k_dist <<<2048, 256, 0, stream>>>(corr, D);
  k_colsum<<<NB,  256, 0, stream>>>(D, P);
  k_rowsum<<<NB,  256, 0, stream>>>(P);
  k_dsum <<<2048, 256, 0, stream>>>(D, P, Ds);
  k_dp   <<<NB,   256, 0, stream>>>(Ds, (float*)d_out);
}